// DirGCNConv_Qin_27152783245349
// MI455X (gfx1250) — compile-verified
//
#include <hip/hip_runtime.h>
#include <hip/hip_bf16.h>

typedef __attribute__((ext_vector_type(2))) float v2f;
typedef __attribute__((ext_vector_type(8))) float v8f;

#define D 128  // D_IN == D_OUT == 128

// ---------------------------------------------------------------------------
// Phase 0: zero the aggregation buffer (float4 stores, fully coalesced)
// ---------------------------------------------------------------------------
__global__ void zero_kernel(float4* __restrict__ p, int n4) {
    int i = blockIdx.x * blockDim.x + threadIdx.x;
    if (i < n4) p[i] = make_float4(0.f, 0.f, 0.f, 0.f);
}

// ---------------------------------------------------------------------------
// Phase 1: edge scatter. One wave32 per edge: each lane loads a float4 of
// x[src] (one global_load_b128) and issues 4 global_atomic_add_f32 into
// agg[dst]. x and agg are L2-resident (51 MB each vs 192 MB L2).
// ---------------------------------------------------------------------------
__global__ void scatter_kernel(const float* __restrict__ x,
                               const int*   __restrict__ ei,   // [2, E] row-major
                               const float* __restrict__ ew,
                               float*       __restrict__ agg,
                               int nEdges) {
    int gid  = blockIdx.x * blockDim.x + threadIdx.x;
    int e    = gid >> 5;
    int lane = gid & 31;
    if (e >= nEdges) return;

    int   dst = ei[e];            // row 0 of edge_index
    int   src = ei[nEdges + e];   // row 1 of edge_index
    float w   = ew[e];

    const float4* xrow = (const float4*)(x + (size_t)src * D);
    float4 v = xrow[lane];        // lane covers floats [4*lane, 4*lane+4)

    float* d = agg + (size_t)dst * D + lane * 4;
    atomicAdd(d + 0, v.x * w);
    atomicAdd(d + 1, v.y * w);
    atomicAdd(d + 2, v.z * w);
    atomicAdd(d + 3, v.w * w);
}

// ---------------------------------------------------------------------------
// Phase 2: out = agg @ W^T + b using V_WMMA_F32_16X16X4_F32 (fp32 parity
// with the reference). Block = 256 threads = 8 waves; block owns a 16-row
// M tile; wave w owns columns [16w, 16w+16). A tile staged in LDS (8 KB),
// B fragments (W columns) preloaded into 64 VGPRs/lane, K=128 in 32 steps.
// ---------------------------------------------------------------------------
__global__ void gemm_wmma_kernel(const float* __restrict__ agg,
                                 const float* __restrict__ W,     // [D_OUT, D_IN] row-major
                                 const float* __restrict__ bias,
                                 float*       __restrict__ out,
                                 int nRows) {
    __shared__ float ldsA[16 * D];   // 8 KB A tile

    int tid    = threadIdx.x;
    int wave   = tid >> 5;
    int lane   = tid & 31;
    int m_base = blockIdx.x * 16;

    // Cooperative A-tile load: 2048 floats = 512 float4, 256 threads -> 2 each.
    {
        const float4* aSrc = (const float4*)(agg + (size_t)m_base * D);
        float4*       aDst = (float4*)ldsA;
        if (m_base + 16 <= nRows) {
            aDst[tid]       = aSrc[tid];
            aDst[tid + 256] = aSrc[tid + 256];
        } else {
            // partial tail tile: guard per row
            for (int t = tid; t < 512; t += 256) {
                int row = t / 32;                      // 32 float4 per row
                aDst[t] = (m_base + row < nRows) ? aSrc[t]
                                                 : make_float4(0.f, 0.f, 0.f, 0.f);
            }
        }
    }
    __syncthreads();

    const int col   = wave * 16 + (lane & 15);         // output column this lane touches
    const int koff  = (lane < 16) ? 0 : 2;             // K sub-offset per ISA A/B layout
    const int row_l = lane & 15;                       // A row within tile for this lane

    // Preload B fragments: B[k][n] = W[n*D + k]; per k-step lane needs
    // W[col*D + 4s + koff .. +1] -> one 8-byte load, 32 steps = 64 VGPRs.
    v2f bfrag[32];
    const float* wcol = W + (size_t)col * D;
#pragma unroll
    for (int s = 0; s < 32; ++s)
        bfrag[s] = *(const v2f*)(wcol + 4 * s + koff);

    v8f c = {};
    const float* arow = ldsA + row_l * D;
#pragma unroll
    for (int s = 0; s < 32; ++s) {
        v2f a = *(const v2f*)(arow + 4 * s + koff);    // ds_load_b64
        c = __builtin_amdgcn_wmma_f32_16x16x4_f32(
                /*neg_a=*/false, a, /*neg_b=*/false, bfrag[s],
                /*c_mod=*/(short)0, c, /*reuse_a=*/false, /*reuse_b=*/false);
    }

    // D layout: VGPR j -> row m_base + j + (lane<16 ? 0 : 8), col = lane%16.
    float bv   = bias[col];
    int   row0 = m_base + ((lane < 16) ? 0 : 8);
#pragma unroll
    for (int j = 0; j < 8; ++j) {
        int row = row0 + j;
        if (row < nRows) out[(size_t)row * D + col] = c[j] + bv;
    }
}

// ---------------------------------------------------------------------------
extern "C" void kernel_launch(void* const* d_in, const int* in_sizes, int n_in,
                              void* d_out, int out_size, void* d_ws, size_t ws_size,
                              hipStream_t stream) {
    const float* x    = (const float*)d_in[0];   // [N, 128]
    const int*   ei   = (const int*)  d_in[1];   // [2, E]
    const float* ew   = (const float*)d_in[2];   // [E]
    const float* W    = (const float*)d_in[3];   // [128, 128]
    const float* bias = (const float*)d_in[4];   // [128]
    float*       out  = (float*)d_out;

    const int nNodes = in_sizes[0] / D;          // 100000
    const int nEdges = in_sizes[2];              // 3200000

    float* agg = (float*)d_ws;                   // [N, 128] fp32 accumulator (51.2 MB)

    // Phase 0: zero accumulator
    {
        int n4     = nNodes * (D / 4);
        int blocks = (n4 + 255) / 256;
        zero_kernel<<<blocks, 256, 0, stream>>>((float4*)agg, n4);
    }
    // Phase 1: edge scatter (one wave per edge)
    {
        long long threads = (long long)nEdges * 32;
        int blocks = (int)((threads + 255) / 256);
        scatter_kernel<<<blocks, 256, 0, stream>>>(x, ei, ew, agg, nEdges);
    }
    // Phase 2: WMMA GEMM + bias
    {
        int mTiles = (nNodes + 15) / 16;         // 6250
        gemm_wmma_kernel<<<mTiles, 256, 0, stream>>>(agg, W, bias, out, nNodes);
    }
}